// GINBackbone_53979148976510
// MI455X (gfx1250) — compile-verified
//
#include <hip/hip_runtime.h>

#define N_NODES 50000
#define N_EDGES 800000
#define HIDDEN  128
#define N_LAYERS 3

typedef float v2f __attribute__((ext_vector_type(2)));
typedef float v8f __attribute__((ext_vector_type(8)));

// ---------------------------------------------------------------------------
// Zero the aggregation buffer (float4 stores).
// ---------------------------------------------------------------------------
__global__ __launch_bounds__(256) void gin_zero_kernel(float* __restrict__ p, int n4) {
    int i = blockIdx.x * blockDim.x + threadIdx.x;
    if (i < n4) {
        ((float4*)p)[i] = make_float4(0.f, 0.f, 0.f, 0.f);
    }
}

// ---------------------------------------------------------------------------
// Edge scatter: agg[dst] += x[src].  8 edges/block, 32 lanes/edge,
// float4 (b128) gather per lane + 4 hardware f32 atomics.
// Feature table (25.6MB) is L2-resident on MI455X (192MB L2).
// ---------------------------------------------------------------------------
__global__ __launch_bounds__(256) void gin_scatter_kernel(
    const float* __restrict__ x,
    const long long* __restrict__ ei,   // [2, E] int64
    float* __restrict__ agg) {
    int e  = blockIdx.x * 8 + (threadIdx.x >> 5);
    int f4 = (threadIdx.x & 31) * 4;
    if (e >= N_EDGES) return;
    int src = (int)ei[e];
    int dst = (int)ei[N_EDGES + e];
    float4 v = *(const float4*)(x + (size_t)src * HIDDEN + f4);
    float* a = agg + (size_t)dst * HIDDEN + f4;
    unsafeAtomicAdd(a + 0, v.x);
    unsafeAtomicAdd(a + 1, v.y);
    unsafeAtomicAdd(a + 2, v.z);
    unsafeAtomicAdd(a + 3, v.w);
}

// ---------------------------------------------------------------------------
// Fused GIN MLP for one layer, one 16-row slab per block:
//   A   = x + agg                       (16 x 128, staged in LDS)
//   Y   = relu(A @ W1 + b1)             (16 x 128, kept in LDS)
//   out = relu(Y @ W2 + b2)             (16 x 128, to global)
// 256 threads = 8 waves; wave w owns output columns [16w, 16w+16).
// K=128 reduction via 32 chained V_WMMA_F32_16X16X4_F32 steps (fp32-exact).
// ---------------------------------------------------------------------------
__global__ __launch_bounds__(256) void gin_mlp_kernel(
    const float* __restrict__ x,
    const float* __restrict__ agg,
    const float* __restrict__ W1, const float* __restrict__ b1,
    const float* __restrict__ W2, const float* __restrict__ b2,
    float* __restrict__ out) {

    __shared__ float Abuf[16 * HIDDEN];   // 8 KB
    __shared__ float Ybuf[16 * HIDDEN];   // 8 KB

    const int row_base = blockIdx.x * 16;
    const int tid  = threadIdx.x;
    const int lane = tid & 31;
    const int wave = tid >> 5;            // 0..7
    const int col_base = wave * 16;
    const int mn = lane & 15;             // row (A frag) / col (B,C frags)
    const int kk = (lane >> 4) * 2;       // K sub-offset: lanes 0-15 -> 0, 16-31 -> 2
    const int half = lane >> 4;           // C/D row group

    // ---- Stage A = x + agg into LDS: 2048 floats, 8 per thread -------------
    {
        int i = tid * 8;
        size_t g = (size_t)row_base * HIDDEN + i;
        float4 xa = *(const float4*)(x   + g);
        float4 ga = *(const float4*)(agg + g);
        float4 xb = *(const float4*)(x   + g + 4);
        float4 gb = *(const float4*)(agg + g + 4);
        *(float4*)(Abuf + i)     = make_float4(xa.x + ga.x, xa.y + ga.y,
                                               xa.z + ga.z, xa.w + ga.w);
        *(float4*)(Abuf + i + 4) = make_float4(xb.x + gb.x, xb.y + gb.y,
                                               xb.z + gb.z, xb.w + gb.w);
    }
    __syncthreads();

    // ---- GEMM1: Y = relu(A @ W1 + b1) --------------------------------------
    {
        float bias = b1[col_base + mn];
        v8f acc;
        #pragma unroll
        for (int i = 0; i < 8; ++i) acc[i] = bias;

        #pragma unroll
        for (int ks = 0; ks < HIDDEN / 4; ++ks) {
            int k = ks * 4 + kk;
            v2f a, b;
            a.x = Abuf[mn * HIDDEN + k];
            a.y = Abuf[mn * HIDDEN + k + 1];
            b.x = W1[(size_t)k       * HIDDEN + col_base + mn];
            b.y = W1[(size_t)(k + 1) * HIDDEN + col_base + mn];
            acc = __builtin_amdgcn_wmma_f32_16x16x4_f32(
                false, a, false, b, (short)0, acc, false, false);
        }
        #pragma unroll
        for (int v = 0; v < 8; ++v) {
            int m = v + 8 * half;
            float val = acc[v];
            Ybuf[m * HIDDEN + col_base + mn] = val > 0.f ? val : 0.f;
        }
    }
    __syncthreads();

    // ---- GEMM2: out = relu(Y @ W2 + b2) ------------------------------------
    {
        float bias = b2[col_base + mn];
        v8f acc;
        #pragma unroll
        for (int i = 0; i < 8; ++i) acc[i] = bias;

        #pragma unroll
        for (int ks = 0; ks < HIDDEN / 4; ++ks) {
            int k = ks * 4 + kk;
            v2f a, b;
            a.x = Ybuf[mn * HIDDEN + k];
            a.y = Ybuf[mn * HIDDEN + k + 1];
            b.x = W2[(size_t)k       * HIDDEN + col_base + mn];
            b.y = W2[(size_t)(k + 1) * HIDDEN + col_base + mn];
            acc = __builtin_amdgcn_wmma_f32_16x16x4_f32(
                false, a, false, b, (short)0, acc, false, false);
        }
        #pragma unroll
        for (int v = 0; v < 8; ++v) {
            int m = v + 8 * half;
            float val = acc[v];
            out[(size_t)(row_base + m) * HIDDEN + col_base + mn] = val > 0.f ? val : 0.f;
        }
    }
}

// ---------------------------------------------------------------------------
// Host-side orchestration (graph-capture safe: stream-only launches).
// ws layout: [agg: N*H floats][xbuf: N*H floats]  (51.2 MB total)
// Ping-pong: L0 feat->d_out, L1 d_out->xbuf, L2 xbuf->d_out.
// ---------------------------------------------------------------------------
extern "C" void kernel_launch(void* const* d_in, const int* in_sizes, int n_in,
                              void* d_out, int out_size, void* d_ws, size_t ws_size,
                              hipStream_t stream) {
    const float*     feat = (const float*)d_in[0];
    const long long* ei   = (const long long*)d_in[1];
    const float*     W1   = (const float*)d_in[2];
    const float*     b1   = (const float*)d_in[3];
    const float*     W2   = (const float*)d_in[4];
    const float*     b2   = (const float*)d_in[5];
    float* out = (float*)d_out;

    float* agg  = (float*)d_ws;
    float* xbuf = agg + (size_t)N_NODES * HIDDEN;

    const int n4        = N_NODES * HIDDEN / 4;
    const int zeroGrid  = (n4 + 255) / 256;
    const int scatGrid  = N_EDGES / 8;       // 100000, exact
    const int mlpGrid   = N_NODES / 16;      // 3125, exact

    const float* xin = feat;
    for (int l = 0; l < N_LAYERS; ++l) {
        float* xout = (l == 1) ? xbuf : out;

        gin_zero_kernel<<<zeroGrid, 256, 0, stream>>>(agg, n4);
        gin_scatter_kernel<<<scatGrid, 256, 0, stream>>>(xin, ei, agg);
        gin_mlp_kernel<<<mlpGrid, 256, 0, stream>>>(
            xin, agg,
            W1 + (size_t)l * HIDDEN * HIDDEN, b1 + (size_t)l * HIDDEN,
            W2 + (size_t)l * HIDDEN * HIDDEN, b2 + (size_t)l * HIDDEN,
            xout);

        xin = xout;
    }
}